// MeanSquaredError2_7541962572203
// MI455X (gfx1250) — compile-verified
//
#include <hip/hip_runtime.h>

#define NJ      14
#define COLS    14
#define BATCH   16384
#define NPAIRS  (BATCH * NJ)            // 229376 (b, j) pairs
#define WPB     8                        // waves (pairs) per block
#define NBLOCKS (NPAIRS / WPB)           // 28672 partial sums
#define NPART   NBLOCKS                  // 28672 = 256 * 112

typedef __attribute__((ext_vector_type(2))) float v2f;
typedef __attribute__((ext_vector_type(8))) float v8f;

// ---------------------------------------------------------------------------
// Kernel 1: one wave32 per (b,j) pair. Streaming float4 loads of the 196
// contiguous heatmap floats, per-lane first-occurrence argmax scan, wave32
// butterfly (max-val, min-idx) reduction, per-pair squared-error contribution,
// deterministic in-block sum of the 8 wave results -> d_ws[blockIdx.x].
// ---------------------------------------------------------------------------
__global__ __launch_bounds__(256) void mse_argmax_partial(
    const float* __restrict__ h,
    const float* __restrict__ t,
    float* __restrict__ part)
{
    const int lane = threadIdx.x & 31;
    const int wave = threadIdx.x >> 5;
    const int pair = blockIdx.x * WPB + wave;     // grid sized exactly: always valid

    const float4* __restrict__ hv =
        (const float4*)(h + (size_t)pair * (COLS * COLS)); // 49 float4s per pair

    // Stream-ahead prefetch into L2 (global_prefetch_b8); clamp to stay mapped.
    {
        int pf = pair + 8 * WPB;
        if (pf >= NPAIRS) pf = NPAIRS - 1;
        __builtin_prefetch(h + (size_t)pf * (COLS * COLS) + lane * 4, 0, 0);
    }

    // Per-lane scan in ascending flat index => strict '>' keeps first occurrence.
    float bestV;
    int   bestI;
    {
        float4 a = hv[lane];                       // indices 4*lane .. 4*lane+3
        int base = 4 * lane;
        bestV = a.x; bestI = base;
        if (a.y > bestV) { bestV = a.y; bestI = base + 1; }
        if (a.z > bestV) { bestV = a.z; bestI = base + 2; }
        if (a.w > bestV) { bestV = a.w; bestI = base + 3; }
    }
    if (lane < 17) {                               // 49 float4s: lanes 0..16 take a 2nd
        float4 a = hv[lane + 32];
        int base = 4 * (lane + 32);
        if (a.x > bestV) { bestV = a.x; bestI = base; }
        if (a.y > bestV) { bestV = a.y; bestI = base + 1; }
        if (a.z > bestV) { bestV = a.z; bestI = base + 2; }
        if (a.w > bestV) { bestV = a.w; bestI = base + 3; }
    }

    // wave32 butterfly: larger value wins; on exact tie, smaller index wins.
    #pragma unroll
    for (int m = 16; m >= 1; m >>= 1) {
        float ov = __shfl_xor(bestV, m, 32);
        int   oi = __shfl_xor(bestI, m, 32);
        if (ov > bestV || (ov == bestV && oi < bestI)) { bestV = ov; bestI = oi; }
    }

    __shared__ float ssum[WPB];
    if (lane == 0) {
        int y = bestI / COLS;
        int x = bestI - y * COLS;
        float px = (float)x * 0.0625f;             // x * COL / 224, exact
        float py = (float)y * 0.0625f;
        int b = pair / NJ;
        int j = pair - b * NJ;
        // hstack-reshape quirk: p[b,j,0]=concat[b,j..] -> compare px with
        // t_flat[b*28 + j] and py with t_flat[b*28 + 14 + j].
        const float* tb = t + (size_t)b * (2 * NJ);
        float dx = px - tb[j];
        float dy = py - tb[NJ + j];
        ssum[wave] = dx * dx + dy * dy;
    }
    __syncthreads();
    if (threadIdx.x == 0) {
        float s = 0.0f;
        #pragma unroll
        for (int w = 0; w < WPB; ++w) s += ssum[w];
        part[blockIdx.x] = s;                      // deterministic order
    }
}

// ---------------------------------------------------------------------------
// Kernel 2: single wave32. Exact fp32 sum of the 28672 partials on the matrix
// pipe: V_WMMA_F32_16X16X4_F32 with all-ones A gives D[m][n] = sum_k B[k][n]
// + C[m][n]; all rows of D are identical, so sum(c[0] over 32 lanes) equals
// 2 * (sum of every value ever loaded into B), independent of B's exact lane
// layout. 4 independent accumulator chains (256 values / iteration) hide the
// C-accumulation latency.
// ---------------------------------------------------------------------------
__global__ __launch_bounds__(32) void wmma_reduce(
    const float* __restrict__ part,
    float* __restrict__ out)
{
    const int lane = threadIdx.x;                  // 0..31, full wave, EXEC all 1s
    v2f a; a[0] = 1.0f; a[1] = 1.0f;               // all-ones A (16x4 f32)
    v8f c0 = {}; v8f c1 = {}; v8f c2 = {}; v8f c3 = {};

    for (int base = 0; base < NPART; base += 256) {
        v2f b0, b1, b2, b3;
        b0[0] = part[base +   0 + lane];  b0[1] = part[base +  32 + lane];
        b1[0] = part[base +  64 + lane];  b1[1] = part[base +  96 + lane];
        b2[0] = part[base + 128 + lane];  b2[1] = part[base + 160 + lane];
        b3[0] = part[base + 192 + lane];  b3[1] = part[base + 224 + lane];
        c0 = __builtin_amdgcn_wmma_f32_16x16x4_f32(false, a, false, b0, (short)0, c0, false, false);
        c1 = __builtin_amdgcn_wmma_f32_16x16x4_f32(false, a, false, b1, (short)0, c1, false, false);
        c2 = __builtin_amdgcn_wmma_f32_16x16x4_f32(false, a, false, b2, (short)0, c2, false, false);
        c3 = __builtin_amdgcn_wmma_f32_16x16x4_f32(false, a, false, b3, (short)0, c3, false, false);
    }

    float s = c0[0] + c1[0] + c2[0] + c3[0];       // row-0/row-8 copies of column sums
    #pragma unroll
    for (int m = 16; m >= 1; m >>= 1) s += __shfl_xor(s, m, 32);
    if (lane == 0) {
        // s == 2 * total (rows duplicated across the two half-wave row groups)
        out[0] = s * 0.5f / (float)NPAIRS;         // n = B*NJ = 229376
    }
}

// ---------------------------------------------------------------------------
// inputs (setup_inputs order): 0=o (unused), 1=h, 2=t, 3=v (unused)
// ---------------------------------------------------------------------------
extern "C" void kernel_launch(void* const* d_in, const int* in_sizes, int n_in,
                              void* d_out, int out_size, void* d_ws, size_t ws_size,
                              hipStream_t stream) {
    const float* h = (const float*)d_in[1];
    const float* t = (const float*)d_in[2];
    float* out  = (float*)d_out;
    float* part = (float*)d_ws;                    // NPART * 4 = 112 KB scratch

    mse_argmax_partial<<<NBLOCKS, 256, 0, stream>>>(h, t, part);
    wmma_reduce<<<1, 32, 0, stream>>>(part, out);
}